// DiffusionPolicy_34746285424703
// MI455X (gfx1250) — compile-verified
//
#include <hip/hip_runtime.h>
#include <math.h>

// ---------------------------------------------------------------------------
// DiffusionPolicy forward for MI455X (gfx1250), f32 WMMA (16x16x4) throughout,
// with async global->LDS staging of GEMM B panels (ASYNCcnt path).
// ---------------------------------------------------------------------------

typedef float v2f __attribute__((ext_vector_type(2)));
typedef float v8f __attribute__((ext_vector_type(8)));

#define NTOK   4096     // B*N = 16*256
#define DMODEL 256
#define NHEADS 8
#define DHEAD  32
#define NBATCH 16
#define NSEQ   256
#define FFDIM  1024

__device__ __forceinline__ v8f wmma4(v2f a, v2f b, v8f c) {
  // D = A(16x4,f32) * B(4x16,f32) + C(16x16,f32); exact f32 on matrix pipes.
  return __builtin_amdgcn_wmma_f32_16x16x4_f32(false, a, false, b, (short)0, c,
                                               false, false);
}

__device__ __forceinline__ float silu_f(float v)  { return v / (1.0f + __expf(-v)); }
__device__ __forceinline__ float gelu_f(float v)  { return 0.5f * v * (1.0f + erff(v * 0.70710678118654752f)); }
__device__ __forceinline__ float leaky_f(float v) { return v > 0.0f ? v : 0.01f * v; }

// async copy 16B from global to LDS (ASYNCcnt-tracked tensor/async data path)
__device__ __forceinline__ void async_copy_b128(uint32_t lds_off, const void* gaddr) {
  asm volatile("global_load_async_to_lds_b128 %0, %1, off"
               :: "v"(lds_off), "v"(gaddr) : "memory");
}
__device__ __forceinline__ void wait_asynccnt0() {
  asm volatile("s_wait_asynccnt 0x0" ::: "memory");
}

// ---------------------------------------------------------------------------
// Fused CNN backbone: 3 valid 3x3 convs (leaky) + lin1 (leaky) + proj -> X.
// One workgroup per token; strip-mined so everything lives in LDS (<50KB).
// ---------------------------------------------------------------------------
__global__ void cnn_kernel(const float* __restrict__ cov,
                           const float* __restrict__ w0, const float* __restrict__ b0,
                           const float* __restrict__ w1, const float* __restrict__ b1,
                           const float* __restrict__ w2, const float* __restrict__ b2,
                           const float* __restrict__ lw, const float* __restrict__ lb,
                           const float* __restrict__ pw, const float* __restrict__ pb,
                           float* __restrict__ X) {
  __shared__ float win[4 * 8 * 32];    // input strip rows y0..y0+7
  __shared__ float a1[32 * 6 * 30];    // conv1 strip (6 rows)
  __shared__ float a2[32 * 4 * 28];    // conv2 strip (4 rows)
  __shared__ float a3[32 * 2 * 26];    // conv3 strip (2 rows)
  __shared__ float red[8 * 32];
  __shared__ float lat[32];

  const int t = threadIdx.x;           // 256 threads
  const size_t tok = blockIdx.x;
  const float* inp = cov + tok * 4 * 1024;

  const int n = t & 31;                // lin1 output channel
  const int g = t >> 5;                // K-slice group
  float acc = 0.0f;                    // lin1 partial

  for (int s = 0; s < 13; ++s) {       // 13 strips of 2 conv3 rows
    const int y0 = 2 * s;
    // stage input rows
    for (int i = t; i < 1024; i += 256) {
      int c = i >> 8, rem = i & 255, yl = rem >> 5, x = rem & 31;
      win[i] = inp[c * 1024 + (y0 + yl) * 32 + x];
    }
    __syncthreads();
    // conv1: 32co x 6row x 30col
    for (int i = t; i < 5760; i += 256) {
      int co = i / 180, rem = i % 180, yl = rem / 30, x = rem % 30;
      float sum = b0[co];
      #pragma unroll
      for (int ci = 0; ci < 4; ++ci)
        #pragma unroll
        for (int ky = 0; ky < 3; ++ky)
          #pragma unroll
          for (int kx = 0; kx < 3; ++kx)
            sum += win[ci * 256 + (yl + ky) * 32 + (x + kx)] *
                   w0[((co * 4 + ci) * 3 + ky) * 3 + kx];
      a1[i] = leaky_f(sum);
    }
    __syncthreads();
    // conv2: 32co x 4row x 28col
    for (int i = t; i < 3584; i += 256) {
      int co = i / 112, rem = i % 112, yl = rem / 28, x = rem % 28;
      float sum = b1[co];
      for (int ci = 0; ci < 32; ++ci)
        #pragma unroll
        for (int ky = 0; ky < 3; ++ky)
          #pragma unroll
          for (int kx = 0; kx < 3; ++kx)
            sum += a1[ci * 180 + (yl + ky) * 30 + (x + kx)] *
                   w1[((co * 32 + ci) * 3 + ky) * 3 + kx];
      a2[i] = leaky_f(sum);
    }
    __syncthreads();
    // conv3: 32co x 2row x 26col
    for (int i = t; i < 1664; i += 256) {
      int co = i / 52, rem = i % 52, yl = rem / 26, x = rem % 26;
      float sum = b2[co];
      for (int ci = 0; ci < 32; ++ci)
        #pragma unroll
        for (int ky = 0; ky < 3; ++ky)
          #pragma unroll
          for (int kx = 0; kx < 3; ++kx)
            sum += a2[ci * 112 + (yl + ky) * 28 + (x + kx)] *
                   w2[((co * 32 + ci) * 3 + ky) * 3 + kx];
      a3[i] = leaky_f(sum);
    }
    __syncthreads();
    // lin1 partial GEMV over this strip (flat order = c*676 + y*26 + x)
    for (int j = 0; j < 208; ++j) {
      int e = g * 208 + j;
      int c = e / 52, rem = e % 52, yl = rem / 26, x = rem % 26;
      size_t flat = (size_t)c * 676 + (size_t)(y0 + yl) * 26 + x;
      acc += a3[e] * lw[flat * 32 + n];   // lanes read consecutive n: coalesced
    }
    __syncthreads();
  }
  red[g * 32 + n] = acc;
  __syncthreads();
  if (t < 32) {
    float s = lb[t];
    #pragma unroll
    for (int gg = 0; gg < 8; ++gg) s += red[gg * 32 + t];
    lat[t] = leaky_f(s);
  }
  __syncthreads();
  // proj: 32 -> 256
  float f = pb[t];
  #pragma unroll
  for (int j = 0; j < 32; ++j) f += lat[j] * pw[j * 256 + t];
  X[tok * DMODEL + t] = f;
}

// ---------------------------------------------------------------------------
// Generic f32 WMMA GEMM: C = act(A[MxK] @ B[KxN] + bias) (+ R residual).
// Block = 256 threads = 8 waves; each wave owns a 32x64 tile of C.
// The 64-wide B panel is shared by all waves: staged chunk-wise (32 k-deep,
// 8 KB) into LDS with global_load_async_to_lds_b128 + s_wait_asynccnt.
// Requires: N % 64 == 0, K % 32 == 0 (true for all call sites).
// ACT: 0 none, 1 silu, 2 gelu.
// ---------------------------------------------------------------------------
template <int ACT, bool RES>
__global__ void gemm_kernel(const float* __restrict__ A, const float* __restrict__ Bw,
                            const float* __restrict__ bias, const float* __restrict__ R,
                            float* __restrict__ C, int M, int K, int N) {
  __shared__ float bs[32 * 64];               // B chunk staging (8 KB)
  const int lane = threadIdx.x & 31;
  const int wave = threadIdx.x >> 5;
  const int row0 = blockIdx.y * 256 + wave * 32;
  const int col0 = blockIdx.x * 64;
  // NOTE: no early exit — all waves participate in async staging + barriers.
  // Tail waves (row0 >= M) compute on clamped rows and mask their stores.

  const int klo = (lane & 16) ? 2 : 0;        // A/B frag K split (ISA 7.12.2)
  const int ll  = lane & 15;

  size_t arow[2];
  #pragma unroll
  for (int mt = 0; mt < 2; ++mt) {
    int m = row0 + mt * 16 + ll; if (m >= M) m = M - 1;
    arow[mt] = (size_t)m * (size_t)K;
  }

  const v8f vzero = {0.f,0.f,0.f,0.f,0.f,0.f,0.f,0.f};
  v8f acc[2][4];
  #pragma unroll
  for (int mt = 0; mt < 2; ++mt)
    #pragma unroll
    for (int nt = 0; nt < 4; ++nt) acc[mt][nt] = vzero;

  // per-thread staging assignment: 512 x 16B per chunk -> 2 per thread
  const uint32_t bs_base = (uint32_t)(uintptr_t)&bs[0];   // LDS byte offset
  const int e0 = threadIdx.x;                 // -> (kk = e>>4, 16B slot = e&15)
  const int e1 = threadIdx.x + 256;

  for (int k0 = 0; k0 < K; k0 += 32) {
    // async-copy B[k0..k0+31][col0..col0+63] into LDS
    {
      const char* g0 = (const char*)(Bw + (size_t)(k0 + (e0 >> 4)) * N + col0) + (e0 & 15) * 16;
      const char* g1 = (const char*)(Bw + (size_t)(k0 + (e1 >> 4)) * N + col0) + (e1 & 15) * 16;
      async_copy_b128(bs_base + (uint32_t)((e0 >> 4) * 256 + (e0 & 15) * 16), g0);
      async_copy_b128(bs_base + (uint32_t)((e1 >> 4) * 256 + (e1 & 15) * 16), g1);
      wait_asynccnt0();
    }
    __syncthreads();

    #pragma unroll
    for (int kk = 0; kk < 32; kk += 4) {
      const int k = k0 + kk;
      v2f af[2], bf[4];
      #pragma unroll
      for (int mt = 0; mt < 2; ++mt) {
        const float* ap = A + arow[mt] + k + klo;
        af[mt].x = ap[0]; af[mt].y = ap[1];
      }
      #pragma unroll
      for (int nt = 0; nt < 4; ++nt) {
        const float* bp = bs + (kk + klo) * 64 + nt * 16 + ll;
        bf[nt].x = bp[0]; bf[nt].y = bp[64];
      }
      #pragma unroll
      for (int mt = 0; mt < 2; ++mt)
        #pragma unroll
        for (int nt = 0; nt < 4; ++nt)
          acc[mt][nt] = wmma4(af[mt], bf[nt], acc[mt][nt]);
    }
    __syncthreads();          // protect bs before next chunk overwrites it
  }

  const int half8 = (lane & 16) ? 8 : 0;
  #pragma unroll
  for (int mt = 0; mt < 2; ++mt)
    #pragma unroll
    for (int nt = 0; nt < 4; ++nt)
      #pragma unroll
      for (int r = 0; r < 8; ++r) {
        int m = row0 + mt * 16 + r + half8;
        int nn = col0 + nt * 16 + ll;
        if (m < M) {
          float v = acc[mt][nt][r] + bias[nn];
          if (ACT == 1) v = silu_f(v);
          else if (ACT == 2) v = gelu_f(v);
          if (RES) v += R[(size_t)m * N + nn];
          C[(size_t)m * N + nn] = v;
        }
      }
}

// ---------------------------------------------------------------------------
// Radius-masked multi-head attention for one (batch, head, 32-row chunk).
// Block = 64 threads = 2 waves; each wave: 16 query rows.
// S = QK^T/sqrt(32) + mask, softmax in registers, P staged in LDS, O = P V.
// ---------------------------------------------------------------------------
__global__ void attn_kernel(const float* __restrict__ qkv,
                            const float* __restrict__ pos,
                            float* __restrict__ out) {
  __shared__ float prob[2 * 16 * 256];   // per-wave probability tiles (32 KB)
  const int lane  = threadIdx.x & 31;
  const int wave  = threadIdx.x >> 5;
  const int blk   = blockIdx.x;
  const int chunk = blk & 7;
  const int h     = (blk >> 3) & 7;
  const int b     = blk >> 6;
  const int m0    = chunk * 32 + wave * 16;    // query tile base (0..240)
  const int klo   = (lane & 16) ? 2 : 0;
  const int lq    = lane & 15;
  const int half8 = (lane & 16) ? 8 : 0;

  const float* Qp = qkv + ((size_t)(b * NSEQ + m0 + lq)) * 768 + h * DHEAD + klo;

  const v8f vzero = {0.f,0.f,0.f,0.f,0.f,0.f,0.f,0.f};
  v8f sc[16];
  #pragma unroll
  for (int tt = 0; tt < 16; ++tt) sc[tt] = vzero;

  // S = Q K^T : 16 col tiles x 8 K-steps
  for (int tt = 0; tt < 16; ++tt) {
    const float* Kp = qkv + ((size_t)(b * NSEQ + tt * 16 + lq)) * 768 + DMODEL + h * DHEAD + klo;
    v8f c = sc[tt];
    #pragma unroll
    for (int k = 0; k < DHEAD; k += 4) {
      v2f av; av.x = Qp[k]; av.y = Qp[k + 1];
      v2f bv; bv.x = Kp[k]; bv.y = Kp[k + 1];
      c = wmma4(av, bv, c);
    }
    sc[tt] = c;
  }

  // scale + radius mask (RADIUS^2 = 0.0625)
  float rpx[8], rpy[8];
  #pragma unroll
  for (int r = 0; r < 8; ++r) {
    int row = m0 + r + half8;
    rpx[r] = pos[((size_t)b * NSEQ + row) * 2];
    rpy[r] = pos[((size_t)b * NSEQ + row) * 2 + 1];
  }
  const float scale = 0.17677669529663689f;   // 1/sqrt(32)
  for (int tt = 0; tt < 16; ++tt) {
    int col = tt * 16 + lq;
    float cx = pos[((size_t)b * NSEQ + col) * 2];
    float cy = pos[((size_t)b * NSEQ + col) * 2 + 1];
    #pragma unroll
    for (int r = 0; r < 8; ++r) {
      float dx = rpx[r] - cx, dy = rpy[r] - cy;
      float v = sc[tt][r] * scale;
      if (dx * dx + dy * dy > 0.0625f) v += -1e9f;
      sc[tt][r] = v;
    }
  }

  // row softmax: in-register over tiles, cross-lane over 16-lane half
  float inv[8];
  #pragma unroll
  for (int r = 0; r < 8; ++r) {
    float mx = -3.0e38f;
    #pragma unroll
    for (int tt = 0; tt < 16; ++tt) mx = fmaxf(mx, sc[tt][r]);
    #pragma unroll
    for (int m = 1; m < 16; m <<= 1) mx = fmaxf(mx, __shfl_xor(mx, m, 32));
    float s = 0.0f;
    #pragma unroll
    for (int tt = 0; tt < 16; ++tt) { float e = __expf(sc[tt][r] - mx); sc[tt][r] = e; s += e; }
    #pragma unroll
    for (int m = 1; m < 16; m <<= 1) s += __shfl_xor(s, m, 32);
    inv[r] = 1.0f / s;
  }

  float* pl = prob + wave * 4096;
  for (int tt = 0; tt < 16; ++tt)
    #pragma unroll
    for (int r = 0; r < 8; ++r)
      pl[(r + half8) * 256 + tt * 16 + lq] = sc[tt][r] * inv[r];

  // O = P V : A-frags from LDS, 2 col tiles (DHEAD=32), 64 K-steps
  v8f oc0 = vzero, oc1 = vzero;
  for (int k = 0; k < NSEQ; k += 4) {
    v2f av; av.x = pl[lq * 256 + k + klo]; av.y = pl[lq * 256 + k + klo + 1];
    const float* Vp = qkv + ((size_t)(b * NSEQ + k + klo)) * 768 + 2 * DMODEL + h * DHEAD;
    v2f b0v; b0v.x = Vp[lq];      b0v.y = Vp[768 + lq];
    v2f b1v; b1v.x = Vp[16 + lq]; b1v.y = Vp[768 + 16 + lq];
    oc0 = wmma4(av, b0v, oc0);
    oc1 = wmma4(av, b1v, oc1);
  }
  #pragma unroll
  for (int r = 0; r < 8; ++r) {
    size_t row = (size_t)b * NSEQ + m0 + r + half8;
    out[row * DMODEL + h * DHEAD + lq]      = oc0[r];
    out[row * DMODEL + h * DHEAD + 16 + lq] = oc1[r];
  }
}

// ---------------------------------------------------------------------------
// LayerNorm over last dim (256): one wave per row, shuffle reductions.
// ---------------------------------------------------------------------------
__global__ void ln_kernel(const float* __restrict__ X, const float* __restrict__ g,
                          const float* __restrict__ bb, float* __restrict__ Y, int M) {
  int row = blockIdx.x * 8 + (threadIdx.x >> 5);
  int lane = threadIdx.x & 31;
  if (row >= M) return;
  const float* x = X + (size_t)row * DMODEL;
  float v[8];
  float s = 0.0f;
  #pragma unroll
  for (int i = 0; i < 8; ++i) { v[i] = x[lane * 8 + i]; s += v[i]; }
  #pragma unroll
  for (int m = 1; m < 32; m <<= 1) s += __shfl_xor(s, m, 32);
  float mean = s * (1.0f / 256.0f);
  float q = 0.0f;
  #pragma unroll
  for (int i = 0; i < 8; ++i) { float d = v[i] - mean; q += d * d; }
  #pragma unroll
  for (int m = 1; m < 32; m <<= 1) q += __shfl_xor(q, m, 32);
  float invs = rsqrtf(q * (1.0f / 256.0f) + 1e-5f);
  #pragma unroll
  for (int i = 0; i < 8; ++i) {
    int c = lane * 8 + i;
    Y[(size_t)row * DMODEL + c] = (v[i] - mean) * invs * g[c] + bb[c];
  }
}

// K=2 input MLP first layer with SiLU: C[m][n] = silu(a0*w[0][n]+a1*w[1][n]+b)
__global__ void gemm_k2_silu(const float* __restrict__ A, const float* __restrict__ W,
                             const float* __restrict__ bias, float* __restrict__ C) {
  int m = blockIdx.x, nn = threadIdx.x;
  float v = A[(size_t)m * 2] * W[nn] + A[(size_t)m * 2 + 1] * W[256 + nn] + bias[nn];
  C[(size_t)m * DMODEL + nn] = silu_f(v);
}

// sinusoidal time features, replicating the reference's freqs expression
__global__ void tsin_kernel(const int* __restrict__ t, float* __restrict__ TS) {
  int b = blockIdx.x, i = threadIdx.x;               // 128 threads
  float freq = __expf(-logf(10000.0f) * ((float)i / (127.0f * 127.0f)));
  float arg = (float)t[b] * freq;
  TS[(size_t)b * DMODEL + i]       = sinf(arg);
  TS[(size_t)b * DMODEL + 128 + i] = cosf(arg);
}

__global__ void add_temb_kernel(float* __restrict__ X, const float* __restrict__ TE) {
  int m = blockIdx.x, d = threadIdx.x;
  X[(size_t)m * DMODEL + d] += TE[(size_t)(m >> 8) * DMODEL + d];
}

// head second layer: N=2
__global__ void head2_kernel(const float* __restrict__ A, const float* __restrict__ W,
                             const float* __restrict__ bias, float* __restrict__ out) {
  int idx = blockIdx.x * blockDim.x + threadIdx.x;   // 8192
  int m = idx >> 1, j = idx & 1;
  const float* a = A + (size_t)m * DMODEL;
  float s = bias[j];
  for (int k = 0; k < DMODEL; ++k) s += a[k] * W[k * 2 + j];
  out[idx] = s;
}

// ---------------------------------------------------------------------------
extern "C" void kernel_launch(void* const* d_in, const int* in_sizes, int n_in,
                              void* d_out, int out_size, void* d_ws, size_t ws_size,
                              hipStream_t stream) {
  const float* cov  = (const float*)d_in[0];
  const float* act  = (const float*)d_in[1];
  const int*   tint = (const int*)  d_in[2];
  const float* pos  = (const float*)d_in[3];
  // params flattened in dict insertion order after the 4 tensor inputs
  #define PRM(i) ((const float*)d_in[4 + (i)])
  // 0..5 conv w/b, 6..7 lin1, 8..9 proj, 10..13 act1/act2, 14..17 pos1/pos2,
  // 18..21 time1/time2, 22+12l layer params, 70..75 head params

  float* W    = (float*)d_ws;
  float* X    = W;                        // token state     4096*256
  float* Hb   = W + 1048576;              // LN output       4096*256
  float* QKV  = W + 2097152;              // qkv             4096*768
  float* FFH  = W + 5242880;              // ff hidden       4096*1024 (also head hidden)
  float* ATT  = W + 9437184;              // attention out   4096*256
  float* TMP  = W + 10485760;             // embedding hidden 4096*256
  float* TSIN = W + 11534336;             // 16*256
  float* TH   = TSIN + 4096;              // 16*256
  float* TEMB = TH + 4096;                // 16*256
  (void)ws_size; (void)n_in; (void)in_sizes; (void)out_size;

  dim3 blk256(256), blk64(64), blk128(128);
  dim3 g_d(DMODEL / 64, NTOK / 256);      // N=256 GEMMs on 4096 rows
  dim3 g_qkv(768 / 64, NTOK / 256);
  dim3 g_ff(FFDIM / 64, NTOK / 256);
  dim3 g_t(DMODEL / 64, 1);               // M=16 GEMMs

  // 1) CNN backbone -> X (= feat)
  cnn_kernel<<<NTOK, blk256, 0, stream>>>(cov, PRM(0), PRM(1), PRM(2), PRM(3),
                                          PRM(4), PRM(5), PRM(6), PRM(7),
                                          PRM(8), PRM(9), X);
  // 2) action embedding: X += silu(act@act1)@act2
  gemm_k2_silu<<<NTOK, blk256, 0, stream>>>(act, PRM(10), PRM(11), TMP);
  gemm_kernel<0, true><<<g_d, blk256, 0, stream>>>(TMP, PRM(12), PRM(13), X, X,
                                                   NTOK, DMODEL, DMODEL);
  // 3) position embedding
  gemm_k2_silu<<<NTOK, blk256, 0, stream>>>(pos, PRM(14), PRM(15), TMP);
  gemm_kernel<0, true><<<g_d, blk256, 0, stream>>>(TMP, PRM(16), PRM(17), X, X,
                                                   NTOK, DMODEL, DMODEL);
  // 4) time embedding (M=16)
  tsin_kernel<<<NBATCH, blk128, 0, stream>>>(tint, TSIN);
  gemm_kernel<1, false><<<g_t, blk256, 0, stream>>>(TSIN, PRM(18), PRM(19), nullptr,
                                                    TH, NBATCH, DMODEL, DMODEL);
  gemm_kernel<0, false><<<g_t, blk256, 0, stream>>>(TH, PRM(20), PRM(21), nullptr,
                                                    TEMB, NBATCH, DMODEL, DMODEL);
  add_temb_kernel<<<NTOK, blk256, 0, stream>>>(X, TEMB);

  // 5) transformer layers
  for (int l = 0; l < 4; ++l) {
    const int pb = 22 + l * 12;
    ln_kernel<<<NTOK / 8, blk256, 0, stream>>>(X, PRM(pb + 4), PRM(pb + 5), Hb, NTOK);
    gemm_kernel<0, false><<<g_qkv, blk256, 0, stream>>>(Hb, PRM(pb + 0), PRM(pb + 1),
                                                        nullptr, QKV, NTOK, DMODEL, 768);
    attn_kernel<<<NBATCH * NHEADS * 8, blk64, 0, stream>>>(QKV, pos, ATT);
    gemm_kernel<0, true><<<g_d, blk256, 0, stream>>>(ATT, PRM(pb + 2), PRM(pb + 3),
                                                     X, X, NTOK, DMODEL, DMODEL);
    ln_kernel<<<NTOK / 8, blk256, 0, stream>>>(X, PRM(pb + 6), PRM(pb + 7), Hb, NTOK);
    gemm_kernel<2, false><<<g_ff, blk256, 0, stream>>>(Hb, PRM(pb + 8), PRM(pb + 9),
                                                       nullptr, FFH, NTOK, DMODEL, FFDIM);
    gemm_kernel<0, true><<<g_d, blk256, 0, stream>>>(FFH, PRM(pb + 10), PRM(pb + 11),
                                                     X, X, NTOK, FFDIM, DMODEL);
  }

  // 6) noise head
  ln_kernel<<<NTOK / 8, blk256, 0, stream>>>(X, PRM(70), PRM(71), Hb, NTOK);
  gemm_kernel<1, false><<<g_d, blk256, 0, stream>>>(Hb, PRM(72), PRM(73), nullptr,
                                                    FFH, NTOK, DMODEL, DMODEL);
  head2_kernel<<<NTOK * 2 / 256, blk256, 0, stream>>>(FFH, PRM(74), PRM(75),
                                                      (float*)d_out);
  #undef PRM
}